// kernel_wrap_layer_20555713479255
// MI455X (gfx1250) — compile-verified
//
#include <hip/hip_runtime.h>

// Problem constants (fixed by the reference).
#define N_  8
#define C_  256
#define H_  128
#define W_  128
#define K_  5
#define PAD_ 2
#define CHUNK_  16                // channel rows per TDM transfer (8 KB)
#define NCHUNK_ (C_ / CHUNK_)     // 16

typedef float        f4  __attribute__((ext_vector_type(4)));
typedef unsigned int v4u __attribute__((ext_vector_type(4)));
typedef int          v8i __attribute__((ext_vector_type(8)));
typedef int          v4i __attribute__((ext_vector_type(4)));

// Low 32 bits of a generic LDS pointer == byte offset into the workgroup's
// LDS allocation (ISA 10.2: the shared aperture base is in the high bits, the LDS offset in the low 32).
__device__ __forceinline__ unsigned lds_byte_off(const void* p) {
  return (unsigned)(unsigned long long)(uintptr_t)p;
}

// Issue one TDM load: 2-D tile of CHUNK_ rows x W_ fp32 elements, row stride
// H_*W_ elements, from global 'gaddr' into LDS at byte offset 'lds_off'.
// D# packing per CDNA5 ISA.
__device__ __forceinline__ void tdm_load_rows(unsigned long long gaddr,
                                              unsigned lds_off) {
  v4u g0;
  g0[0] = 1u;                                      // count=1 (valid), is_restore=0, gather off
  g0[1] = lds_off;                                 // lds_addr (bytes)
  g0[2] = (unsigned)(gaddr & 0xFFFFFFFFull);       // global_addr[31:0]
  g0[3] = (unsigned)((gaddr >> 32) & 0x01FFFFFFull)// global_addr[56:32]
        | 0x80000000u;                             // type=2 ("image") in bits[127:126]

  v8i g1;
  g1[0] = 0x00020000;            // workgroup_mask=0 (not in cluster), data_size=2 (4B)
  g1[1] = (int)(((unsigned)W_ & 0xFFFFu) << 16);   // tensor_dim0[15:0]=128
  g1[2] = (int)(((unsigned)CHUNK_) << 16);         // tensor_dim0 hi=0 | tensor_dim1[15:0]=16
  g1[3] = (int)(((unsigned)W_) << 16);             // tensor_dim1 hi=0 | tile_dim0=128
  g1[4] = CHUNK_;                                  // tile_dim1=16, tile_dim2=0
  g1[5] = H_ * W_;                                 // tensor_dim0_stride low32 = 16384
  g1[6] = 0;                                       // stride hi | tensor_dim1_stride lo16
  g1[7] = 0;

  v4i z4 = {0, 0, 0, 0};                           // groups 2/3 unused (<=2D tensor)
  v8i z8 = {0, 0, 0, 0, 0, 0, 0, 0};               // 6-arg form: extra group (unused)
  __builtin_amdgcn_tensor_load_to_lds(g0, g1, z4, z4, z8, /*cpol: TH=NT*/ 1);
}

__global__ __launch_bounds__(256)
void scale_fold_kernel(const float* __restrict__ kern,
                       const float* __restrict__ low,
                       float* __restrict__ out) {
  __shared__ __align__(16) float sS[W_];
  __shared__ __align__(16) float sBuf[2][CHUNK_ * W_];   // 2 x 8 KB

  const int y    = blockIdx.x;          // 0..127
  const int n    = blockIdx.y;          // 0..7
  const int t    = threadIdx.x;         // 0..255
  const int wave = t >> 5;
  const int r    = t >> 5;              // channel-row within half-chunk (0..7)
  const int x4   = t & 31;              // float4 column (0..31)

  // Base global address of low_fea[n, 0, y, 0].
  const unsigned long long lowBase =
      (unsigned long long)(uintptr_t)low +
      (unsigned long long)(((size_t)n * C_) * H_ + (size_t)y) * W_ * sizeof(float);

  // Prefetch chunk 0 into buffer 0 (TDM ignores EXEC; issue from wave 0 only).
  if (wave == 0) {
    tdm_load_rows(lowBase, lds_byte_off(&sBuf[0][0]));
  }

  // ---- Phase 1: S[x] = sum_{a,b} kernel[n, a*5+b, y+2-a, x+2-b] ----
  if (t < W_) {
    const int x = t;
    float acc = 0.0f;
#pragma unroll
    for (int a = 0; a < K_; ++a) {
      const int yy = y + PAD_ - a;
      if (yy >= 0 && yy < H_) {
        const float* kp = kern +
            (((size_t)n * (K_ * K_) + (size_t)a * K_) * H_ + (size_t)yy) * W_;
#pragma unroll
        for (int b = 0; b < K_; ++b) {
          const int xx = x + PAD_ - b;
          if (xx >= 0 && xx < W_) acc += kp[(size_t)b * H_ * W_ + xx];
        }
      }
    }
    sS[x] = acc;
  }
  __syncthreads();                       // S visible to all waves

  const f4 sv = ((const f4*)sS)[x4];     // broadcast scale for this column group

  // ---- Phase 2: pipelined TDM stream, ONE barrier per iteration ----
  // iter i:  wait(chunk i) -> barrier (publishes chunk i AND retires reads of
  //          the buffer chunk i+1 overwrites) -> issue chunk i+1 -> consume i.
  for (int i = 0; i < NCHUNK_; ++i) {
    if (wave == 0) {
      __builtin_amdgcn_s_wait_tensorcnt(0);        // chunk i landed in LDS
    }
    __syncthreads();

    if (wave == 0 && (i + 1) < NCHUNK_) {
      const unsigned long long ga =
          lowBase + (unsigned long long)(i + 1) * CHUNK_ * H_ * W_ * sizeof(float);
      tdm_load_rows(ga, lds_byte_off(&sBuf[(i + 1) & 1][0]));   // overlaps consume
    }

    // Consume chunk i: 2 float4 per thread (rows r and r+8 of the chunk).
    const f4* buf = (const f4*)&sBuf[i & 1][0];
    const f4 v0 = buf[t];                // row r,     col x4
    const f4 v1 = buf[t + 256];          // row r + 8, col x4
    const f4 o0 = v0 * sv;
    const f4 o1 = v1 * sv;

    const int c0 = i * CHUNK_ + r;       // channel of v0
    f4* outp0 = (f4*)out + ((((size_t)n * C_ + c0)     * H_ + y) * (W_ / 4) + x4);
    f4* outp1 = (f4*)out + ((((size_t)n * C_ + c0 + 8) * H_ + y) * (W_ / 4) + x4);
    __builtin_nontemporal_store(o0, outp0);        // NT global_store_b128
    __builtin_nontemporal_store(o1, outp1);
  }
}

extern "C" void kernel_launch(void* const* d_in, const int* in_sizes, int n_in,
                              void* d_out, int out_size, void* d_ws, size_t ws_size,
                              hipStream_t stream) {
  (void)in_sizes; (void)n_in; (void)out_size; (void)d_ws; (void)ws_size;
  const float* kern = (const float*)d_in[0];   // (8, 25, 128, 128) fp32
  const float* low  = (const float*)d_in[1];   // (8, 256, 128, 128) fp32
  float* out        = (float*)d_out;           // (8, 256, 128, 128) fp32

  dim3 grid(H_, N_);                            // 1024 blocks
  scale_fold_kernel<<<grid, 256, 0, stream>>>(kern, low, out);
}